// lmrk_discriminator_8443905704057
// MI455X (gfx1250) — compile-verified
//
#include <hip/hip_runtime.h>

// One workgroup (256 threads = 8 wave32) per graph; all activations live in LDS.
// GEMMs: V_WMMA_F32_16X16X4_F32 (full fp32). Weight matrices are DMA'd into LDS
// by the Tensor Data Mover (tensor_load_to_lds) with hardware row padding to the
// bank-conflict-free 136-float stride, double-buffered and overlapped with the
// LDS scatter-add aggregation. Consumption gated by s_wait_tensorcnt + barrier.

#define NP   68      // nodes per graph
#define MP   80      // padded to 5 M-tiles of 16
#define FH   128     // hidden features
#define HS   132     // H/AGG row stride (floats): 132 mod 64 banks = 4 -> conflict-free A frags
#define WSTR 136     // W row stride (floats): 2*136 mod 64 = 16 -> conflict-free B frags
#define EPG  544     // edges per graph
#define NT   256     // threads per block

typedef float v2f __attribute__((ext_vector_type(2)));
typedef float v8f __attribute__((ext_vector_type(8)));
typedef unsigned int u32x4 __attribute__((ext_vector_type(4)));
typedef int i32x4 __attribute__((ext_vector_type(4)));
typedef int i32x8 __attribute__((ext_vector_type(8)));

// dynamic-LDS layout (float offsets)
#define OFF_H    0
#define OFF_AGG  (OFF_H   + MP*HS)      // 10560
#define OFF_WLA  (OFF_AGG + MP*HS)      // 21120 : W_rel  staged by TDM, stride 136
#define OFF_WLB  (OFF_WLA + FH*WSTR)    // 38528 : W_root staged by TDM, stride 136
#define OFF_BIAS (OFF_WLB + FH*WSTR)    // 55936
#define OFF_W1   (OFF_BIAS+ FH)         // 56064 : [rel1 k0 | rel1 k1 | root1 k0 | root1 k1]
#define OFF_SM   (OFF_W1  + 4*FH)       // 56576 : softmax(s)  [68][8]
#define OFF_EDG  (OFF_SM  + NP*8)       // 57120 : packed (dst<<16)|src per edge
#define OFF_RED  (OFF_EDG + EPG)        // 57664
#define SMEM_FLOATS (OFF_RED + FH)      // 57792 floats = 231168 bytes (< 320KB/WGP)

// ---- Tensor Data Mover: DMA a 128x128 fp32 row-major matrix from global into
// LDS, padding each 128-dword row by 8 dwords -> LDS row stride = 136 floats.
// D# packing per CDNA5 ISA (2D tensor, groups 2/3 disabled).
__device__ __forceinline__ void tdm_load_w128(const float* __restrict__ gsrc,
                                              float* ldst)
{
  const unsigned long long ga = (unsigned long long)(uintptr_t)gsrc;
  const unsigned int       la = (unsigned int)(uintptr_t)ldst;  // flat LDS addr: low 32 bits = LDS byte offset

  u32x4 g0;
  g0.x = 1u;                                                   // count=1, user descriptor
  g0.y = la;                                                   // lds_addr (bytes)
  g0.z = (unsigned int)ga;                                     // global_addr[31:0]
  g0.w = (unsigned int)((ga >> 32) & 0x01FFFFFFu) | (2u << 30); // global_addr[56:32] | type=2

  i32x8 g1;
  g1[0] = (2 << 16)      // data_size = 4 bytes
        | (1 << 20)      // pad_enable
        | (6 << 22)      // pad_interval: every 128 dwords (512B row)
        | (7 << 25);     // pad_amount: 8 dwords -> row stride 136 floats
  g1[1] = (int)(128u << 16);  // tensor_dim0[15:0]=128 (bits 79:48 span words 1/2)
  g1[2] = (int)(128u << 16);  // tensor_dim0 hi=0 | tensor_dim1[15:0]=128
  g1[3] = (int)(128u << 16);  // tensor_dim1 hi=0 | tile_dim0=128
  g1[4] = 128;                // tile_dim1=128, tile_dim2=0 (2D)
  g1[5] = 128;                // tensor_dim0_stride = 128 elements
  g1[6] = 0;                  // stride hi / tensor_dim1_stride (unused, 2D tile)
  g1[7] = 0;

  const i32x4 z4 = {0, 0, 0, 0};
#if defined(__clang_major__) && (__clang_major__ >= 23)
  const i32x8 z8 = {0, 0, 0, 0, 0, 0, 0, 0};
  __builtin_amdgcn_tensor_load_to_lds(g0, g1, z4, z4, z8, 0);
#else
  __builtin_amdgcn_tensor_load_to_lds(g0, g1, z4, z4, 0);
#endif
}

// acc[mt] += A[mt-tile, k..k+3] * B[k..k+3, nt-tile] over full K=128, fp32 WMMA.
// A: lane<16 holds row M=lane, K pair (k,k+1); lane>=16 K pair (k+2,k+3).
// B: VGPR0 = B[k+kh][n], VGPR1 = B[k+kh+1][n], kh = 0|2 by lane half.
__device__ __forceinline__ void gemm_acc_f32(v8f acc[5],
                                             const float* __restrict__ A,
                                             const float* __restrict__ Bm,
                                             int lane, int nt)
{
  const int mrow  = lane & 15;
  const int khalf = (lane >> 4) << 1;        // 0 or 2
  const int ncol  = nt * 16 + mrow;
  #pragma unroll 4
  for (int k = 0; k < FH; k += 4) {
    v2f bfrag;
    bfrag.x = Bm[(k + khalf    ) * WSTR + ncol];
    bfrag.y = Bm[(k + khalf + 1) * WSTR + ncol];
    #pragma unroll
    for (int mt = 0; mt < 5; ++mt) {
      const v2f afrag = *(const v2f*)&A[(mt * 16 + mrow) * HS + k + khalf];
      acc[mt] = __builtin_amdgcn_wmma_f32_16x16x4_f32(
          false, afrag, false, bfrag, (short)0, acc[mt], false, false);
    }
  }
}

__global__ __launch_bounds__(NT) void lmrk_gnn_fused(
    const float* __restrict__ x, const long long* __restrict__ ei,
    const float* __restrict__ s,
    const float* __restrict__ Wroot1, const float* __restrict__ Wrel1, const float* __restrict__ b1,
    const float* __restrict__ Wroot2, const float* __restrict__ Wrel2, const float* __restrict__ b2,
    const float* __restrict__ Wroot3, const float* __restrict__ Wrel3, const float* __restrict__ b3,
    const float* __restrict__ fcw, const float* __restrict__ fcb,
    float* __restrict__ out, long long eTot)
{
  extern __shared__ float smem[];
  float* H    = smem + OFF_H;
  float* AGG  = smem + OFF_AGG;
  float* WLA  = smem + OFF_WLA;
  float* WLB  = smem + OFF_WLB;
  float* BIAS = smem + OFF_BIAS;
  float* W1   = smem + OFF_W1;
  float* SM   = smem + OFF_SM;
  int*   EDG  = (int*)(smem + OFF_EDG);
  float* RED  = smem + OFF_RED;

  const int b    = blockIdx.x;
  const int tid  = threadIdx.x;
  const int lane = tid & 31;
  const int wave = tid >> 5;

  // ---- kick off layer-2 weight DMAs immediately (overlap with layer 1)
  if (tid == 0) {
    tdm_load_w128(Wrel2,  WLA);
    tdm_load_w128(Wroot2, WLB);
  }

  // ---- init: zero H/AGG, stage edges as local-id pairs, layer-1 weights, x
  for (int i = tid; i < MP * HS; i += NT) { H[i] = 0.f; AGG[i] = 0.f; }
  for (int e = tid; e < EPG; e += NT) {
    const long long base = (long long)b * EPG + e;
    const int srcl = (int)(ei[base]        - (long long)b * NP);
    const int dstl = (int)(ei[eTot + base] - (long long)b * NP);
    EDG[e] = (dstl << 16) | srcl;
  }
  if (tid < FH) {
    W1[tid]          = Wrel1[tid];
    W1[FH + tid]     = Wrel1[FH + tid];
    W1[2*FH + tid]   = Wroot1[tid];
    W1[3*FH + tid]   = Wroot1[FH + tid];
    BIAS[tid]        = b1[tid];
  }
  if (tid < NP) {   // stash x in padding columns 128/129 of H
    H[tid*HS + 128] = x[((long long)b*NP + tid)*2 + 0];
    H[tid*HS + 129] = x[((long long)b*NP + tid)*2 + 1];
  }
  __syncthreads();

  // ---- layer 1 (F_IN = 2, pure VALU)
  for (int e = tid; e < EPG; e += NT) {
    const int sd = EDG[e];
    const int srcl = sd & 0xffff, dstl = sd >> 16;
    atomicAdd(&AGG[dstl*HS + 0], H[srcl*HS + 128]);
    atomicAdd(&AGG[dstl*HS + 1], H[srcl*HS + 129]);
  }
  __syncthreads();
  for (int i = tid; i < NP * FH; i += NT) {
    const int n = i >> 7, f = i & 127;
    const float a0 = AGG[n*HS],       a1 = AGG[n*HS + 1];
    const float x0 = H[n*HS + 128],   x1 = H[n*HS + 129];
    const float v = BIAS[f] + a0*W1[f] + a1*W1[FH+f] + x0*W1[2*FH+f] + x1*W1[3*FH+f];
    H[n*HS + f] = fmaxf(v, 0.f);
  }
  __syncthreads();

  // ---- layers 2 & 3: LDS scatter-add + two WMMA GEMMs, weights via TDM
  for (int layer = 0; layer < 2; ++layer) {
    const float* bb = layer ? b3 : b2;

    for (int i = tid; i < MP * HS; i += NT) AGG[i] = 0.f;
    if (tid < FH) BIAS[tid] = bb[tid];
    __syncthreads();

    // aggregation overlaps the in-flight weight DMAs; work item = (edge, 4-feat chunk)
    for (int idx = tid; idx < EPG * 32; idx += NT) {
      const int e = idx >> 5, fq = (idx & 31) << 2;
      const int sd = EDG[e];
      const int srcl = sd & 0xffff, dstl = sd >> 16;
      const float4 v = *(const float4*)&H[srcl*HS + fq];
      atomicAdd(&AGG[dstl*HS + fq + 0], v.x);
      atomicAdd(&AGG[dstl*HS + fq + 1], v.y);
      atomicAdd(&AGG[dstl*HS + fq + 2], v.z);
      atomicAdd(&AGG[dstl*HS + fq + 3], v.w);
    }
    if (tid == 0) __builtin_amdgcn_s_wait_tensorcnt(0);  // weights landed in LDS
    __syncthreads();

    v8f acc[5];
    #pragma unroll
    for (int mt = 0; mt < 5; ++mt) acc[mt] = (v8f){0.f,0.f,0.f,0.f,0.f,0.f,0.f,0.f};

    gemm_acc_f32(acc, AGG, WLA, lane, wave);   // agg @ W_rel   (no mid-layer barrier:
    gemm_acc_f32(acc, H,   WLB, lane, wave);   // h   @ W_root   disjoint LDS buffers)
    __syncthreads();                           // all reads of H/WLA/WLB complete

    // prefetch layer-3 weights while we write back and re-aggregate
    if (layer == 0 && tid == 0) {
      tdm_load_w128(Wrel3,  WLA);
      tdm_load_w128(Wroot3, WLB);
    }

    // writeback relu(acc + bias) into H (C/D layout: VGPR r -> row r (+8 for hi lanes))
    const int ncol  = wave * 16 + (lane & 15);
    const float bia = BIAS[ncol];
    const int mbase = (lane >> 4) << 3;
    #pragma unroll
    for (int mt = 0; mt < 5; ++mt) {
      #pragma unroll
      for (int r = 0; r < 8; ++r)
        H[(mt*16 + mbase + r)*HS + ncol] = fmaxf(acc[mt][r] + bia, 0.f);
    }
    __syncthreads();
  }

  // ---- diff-pool: row softmax of s, then xp = s_smᵀ @ h3  ([8,128] per graph)
  if (tid < NP) {
    const float* srow = s + ((long long)b*NP + tid)*8;
    float v[8], mx = -1e30f;
    #pragma unroll
    for (int j = 0; j < 8; ++j) { v[j] = srow[j]; mx = fmaxf(mx, v[j]); }
    float sum = 0.f;
    #pragma unroll
    for (int j = 0; j < 8; ++j) { v[j] = __expf(v[j] - mx); sum += v[j]; }
    const float inv = 1.f / sum;
    #pragma unroll
    for (int j = 0; j < 8; ++j) SM[tid*8 + j] = v[j] * inv;
  }
  __syncthreads();

  {
    const int c  = tid >> 5;            // cluster 0..7
    const int f0 = (tid & 31) << 2;     // 4 features per thread
    float4 p; p.x = p.y = p.z = p.w = 0.f;
    for (int n = 0; n < NP; ++n) {
      const float sw = SM[n*8 + c];
      const float4 hv = *(const float4*)&H[n*HS + f0];
      p.x += sw*hv.x; p.y += sw*hv.y; p.z += sw*hv.z; p.w += sw*hv.w;
    }
    *(float4*)&AGG[c*HS + f0] = p;      // AGG reused as XP[8][128]
  }
  __syncthreads();

  // ---- maxpool((1,8)) + fc
  if (tid < FH) {
    const int c = tid >> 4, j = tid & 15;   // flat index = c*16 + j = tid
    float m = AGG[c*HS + j*8];
    #pragma unroll
    for (int k = 1; k < 8; ++k) m = fmaxf(m, AGG[c*HS + j*8 + k]);
    RED[tid] = m * fcw[tid];
  }
  __syncthreads();
  for (int off = 64; off > 0; off >>= 1) {
    if (tid < off) RED[tid] += RED[tid + off];
    __syncthreads();
  }
  if (tid == 0) out[b] = RED[0] + fcb[0];
}

extern "C" void kernel_launch(void* const* d_in, const int* in_sizes, int n_in,
                              void* d_out, int out_size, void* d_ws, size_t ws_size,
                              hipStream_t stream) {
  const float*     x      = (const float*)d_in[0];
  const long long* ei     = (const long long*)d_in[1];
  // d_in[2] = adj : unused (its contribution is discarded by the reference)
  const float*     s      = (const float*)d_in[3];
  const float*     Wroot1 = (const float*)d_in[4];
  const float*     Wrel1  = (const float*)d_in[5];
  const float*     b1     = (const float*)d_in[6];
  const float*     Wroot2 = (const float*)d_in[7];
  const float*     Wrel2  = (const float*)d_in[8];
  const float*     b2     = (const float*)d_in[9];
  const float*     Wroot3 = (const float*)d_in[10];
  const float*     Wrel3  = (const float*)d_in[11];
  const float*     b3     = (const float*)d_in[12];
  const float*     fcw    = (const float*)d_in[13];
  const float*     fcb    = (const float*)d_in[14];
  float*           out    = (float*)d_out;

  const long long eTot   = (long long)in_sizes[1] / 2;  // total edges
  const int       nGraph = out_size;                    // B = 2048
  const size_t    smem   = (size_t)SMEM_FLOATS * sizeof(float);

  lmrk_gnn_fused<<<dim3(nGraph), dim3(NT), smem, stream>>>(
      x, ei, s, Wroot1, Wrel1, b1, Wroot2, Wrel2, b2,
      Wroot3, Wrel3, b3, fcw, fcb, out, eTot);
}